// ESRNN_14456859918541
// MI455X (gfx1250) — compile-verified
//
#include <hip/hip_runtime.h>
#include <stdint.h>

typedef uint32_t u32;
typedef uint64_t u64;
typedef __attribute__((ext_vector_type(4))) u32 u32x4;
typedef __attribute__((ext_vector_type(4))) int i32x4;
typedef __attribute__((ext_vector_type(8))) int i32x8;

// ---- problem constants ----
#define HW_B 8192
#define HW_T 1024
#define HW_P 12
#define TC   64             // timesteps per chunk
#define NC   (HW_T / TC)    // 16 chunks
#define ROWS 32             // series per block == lanes per wave

// ---- LDS layout (float indices; dynamic LDS assumed to start at offset 0) ----
#define YIN_STRIDE  65      // 64 data + 1 pad dword inserted by TDM pad logic -> bank-conflict-free
#define YOUT_STRIDE 66      // 8B-aligned rows for b64 async stores; 2l mod 64 conflict-free
#define YIN0   0
#define YIN1   (YIN0  + ROWS * YIN_STRIDE)    // 2080
#define YOUT0  (YIN1  + ROWS * YIN_STRIDE)    // 4160
#define YOUT1  (YOUT0 + ROWS * YOUT_STRIDE)   // 6272
#define SBUF   (YOUT1 + ROWS * YOUT_STRIDE)   // 8384
#define SSTRIDE 13                            // 13l mod 64 conflict-free
#define LDS_FLOATS (SBUF + ROWS * SSTRIDE)    // 8800
#define LDS_BYTES  (LDS_FLOATS * 4)           // 35200 bytes

// ---- feature detection ----
#if defined(__has_builtin)
#  if __has_builtin(__builtin_amdgcn_tensor_load_to_lds)
#    define HAVE_TDM_BUILTIN 1
#  else
#    define HAVE_TDM_BUILTIN 0
#  endif
#  if __has_builtin(__builtin_amdgcn_s_wait_asynccnt)
#    define WAIT_ASYNC_LE(n) __builtin_amdgcn_s_wait_asynccnt(n)
#  else
#    define WAIT_ASYNC_LE(n) asm volatile("s_wait_asynccnt %0" :: "i"(n) : "memory")
#  endif
#else
#  define HAVE_TDM_BUILTIN 0
#  define WAIT_ASYNC_LE(n) asm volatile("s_wait_asynccnt %0" :: "i"(n) : "memory")
#endif

#if __has_include(<hip/amd_detail/amd_gfx1250_TDM.h>)
#  define TDM_SIX_ARG 1
#else
#  define TDM_SIX_ARG 0
#endif

// Issue one TDM 2-D tile load: 32 rows x 64 f32, row stride HW_T elements,
// LDS pad of 1 dword every 64 dwords -> LDS row stride 65 dwords.
__device__ __forceinline__ void tdm_load_tile(u64 gaddr, u32 lds_bytes) {
  u32x4 g0;
  g0.x = 0x1u;                                              // count=1, user descriptor
  g0.y = lds_bytes;                                         // LDS byte address of tile
  g0.z = (u32)gaddr;                                        // global_addr[31:0]
  g0.w = ((u32)(gaddr >> 32) & 0x01FFFFFFu) | (2u << 30);   // global_addr[56:32] | type=2

  const u32 TD = 0x40000000u;  // huge tensor_dim0/1 so OOB never triggers
  u64 q0 = (2ull << 16)                       // data_size = 4B
         | (1ull << 20)                       // pad_enable
         | (5ull << 22)                       // pad_interval: every 64 dwords
         | (0ull << 25)                       // pad_amount: 1 dword
         | ((u64)(TD & 0xFFFFu) << 48);       // tensor_dim0[15:0]
  u64 q1 = (u64)(TD >> 16)                    // tensor_dim0[31:16]
         | ((u64)TD << 16)                    // tensor_dim1
         | ((u64)TC << 48);                   // tile_dim0 = 64 elements
  u64 q2 = (u64)ROWS                          // tile_dim1 = 32 rows (tile_dim2 = 0)
         | ((u64)(u32)HW_T << 32);            // tensor_dim0_stride[31:0] = 1024
  u64 q3 = 0;                                 // stride0 hi16 = 0, tensor_dim1_stride = 0

  i32x8 g1;
  g1[0] = (int)(u32)q0; g1[1] = (int)(u32)(q0 >> 32);
  g1[2] = (int)(u32)q1; g1[3] = (int)(u32)(q1 >> 32);
  g1[4] = (int)(u32)q2; g1[5] = (int)(u32)(q2 >> 32);
  g1[6] = (int)(u32)q3; g1[7] = (int)(u32)(q3 >> 32);

#if HAVE_TDM_BUILTIN
  i32x4 z4 = {0, 0, 0, 0};
#  if TDM_SIX_ARG
  i32x8 z8 = {0, 0, 0, 0, 0, 0, 0, 0};
  __builtin_amdgcn_tensor_load_to_lds(g0, g1, z4, z4, z8, 0);
#  else
  __builtin_amdgcn_tensor_load_to_lds(g0, g1, z4, z4, 0);
#  endif
#else
  asm volatile("tensor_load_to_lds %0, %1" :: "s"(g0), "s"(g1) : "memory");
#endif
}

__device__ __forceinline__ float fast_rcp(float x) {
  float r = __builtin_amdgcn_rcpf(x);
  r = r * (2.0f - x * r);   // one Newton iteration -> ~full f32 precision
  return r;
}

extern "C" __global__ __launch_bounds__(ROWS)
void esrnn_hw_scan_gfx1250(const float* __restrict__ y,
                           const float* __restrict__ l0,
                           const float* __restrict__ b0,
                           const float* __restrict__ s0,
                           const float* __restrict__ pAlpha,
                           const float* __restrict__ pBeta,
                           const float* __restrict__ pPhi,
                           const float* __restrict__ pGamma,
                           float* __restrict__ out)
{
  extern __shared__ float smem[];
  const int lane = threadIdx.x;         // 0..31, one series per lane
  const int sb   = blockIdx.x * ROWS;   // first series of this block
  const int gb   = sb + lane;

  const float alpha = pAlpha[0], beta = pBeta[0], phi = pPhi[0], gamma = pGamma[0];
  const float ia = 1.0f - alpha, ibet = 1.0f - beta, ig = 1.0f - gamma;

  float l = l0[gb];
  float b = b0[gb];
  float* srow = smem + SBUF + lane * SSTRIDE;
#pragma unroll
  for (int j = 0; j < HW_P; ++j)
    srow[j] = s0[(size_t)gb * HW_P + j];
  int head = 0;

  // prefetch chunk 0
  tdm_load_tile((u64)(uintptr_t)(y + (size_t)sb * HW_T), (u32)(YIN0 * 4));

  for (int k = 0; k < NC; ++k) {
    // prefetch chunk k+1 into the other input buffer, then wait for chunk k
    if (k + 1 < NC) {
      u32 dst = (u32)((((k + 1) & 1) ? YIN1 : YIN0) * 4);
      tdm_load_tile((u64)(uintptr_t)(y + (size_t)sb * HW_T + (size_t)(k + 1) * TC), dst);
      __builtin_amdgcn_s_wait_tensorcnt(1);   // T_k done, T_{k+1} in flight
    } else {
      __builtin_amdgcn_s_wait_tensorcnt(0);   // last chunk: T_k done
    }
    // ensure stores S_{k-2} (which read outtile[k%2]) retired; S_{k-1} may remain
    WAIT_ASYNC_LE(32);
    asm volatile("" ::: "memory");

    const int ibase = (k & 1) ? YIN1 : YIN0;
    const int obase = (k & 1) ? YOUT1 : YOUT0;
    float* tin  = smem + ibase + lane * YIN_STRIDE;   // 65l mod 64 = l  -> conflict-free
    float* tout = smem + obase + lane * YOUT_STRIDE;  // 66l mod 64 = 2l -> conflict-free

#pragma unroll 8
    for (int t = 0; t < TC; ++t) {
      float yt   = tin[t];
      float sp   = srow[head];                               // s_prev[0]
      float lnew = alpha * yt * fast_rcp(sp) + ia * (l + phi * b);
      float bnew = beta * (lnew - l) + ibet * b * phi;
      float c    = gamma * yt * fast_rcp(lnew) + ig * sp;
      srow[head] = c;                                        // becomes new s[P-1]
      head = (head + 1 == HW_P) ? 0 : head + 1;
      float sn   = srow[head];                               // new s[0] = old s_prev[1]
      tout[t]    = (lnew + bnew * phi) * sn;
      l = lnew; b = bnew;
    }

    // DS writes must be visible before the async engine reads LDS
    asm volatile("s_wait_dscnt 0x0" ::: "memory");

    // drain out tile: one b64 async store per row (32 ops), coalesced 256B per wave
#pragma unroll
    for (int j = 0; j < ROWS; ++j) {
      u32 lds_off = (u32)((obase + j * YOUT_STRIDE + 2 * lane) * 4);
      u64 ga = (u64)(uintptr_t)(out + (size_t)(sb + j) * HW_T + (size_t)k * TC + 2 * lane);
      asm volatile("global_store_async_from_lds_b64 %0, %1, off"
                   :: "v"(ga), "v"(lds_off) : "memory");
    }
  }

  WAIT_ASYNC_LE(0);  // flush all outstanding async stores
}

extern "C" void kernel_launch(void* const* d_in, const int* in_sizes, int n_in,
                              void* d_out, int out_size, void* d_ws, size_t ws_size,
                              hipStream_t stream) {
  (void)in_sizes; (void)n_in; (void)out_size; (void)d_ws; (void)ws_size;
  const float* y     = (const float*)d_in[0];
  const float* l0    = (const float*)d_in[1];
  const float* b0    = (const float*)d_in[2];
  const float* s0    = (const float*)d_in[3];
  const float* alpha = (const float*)d_in[4];
  const float* beta  = (const float*)d_in[5];
  const float* phi   = (const float*)d_in[6];
  const float* gamma = (const float*)d_in[7];
  float* out = (float*)d_out;

  dim3 grid(HW_B / ROWS);   // 256 single-wave workgroups -> spread across WGPs
  dim3 block(ROWS);         // wave32
  esrnn_hw_scan_gfx1250<<<grid, block, LDS_BYTES, stream>>>(
      y, l0, b0, s0, alpha, beta, phi, gamma, out);
}